// CrossEntropyLoss_38749194945159
// MI455X (gfx1250) — compile-verified
//
#include <hip/hip_runtime.h>

#define C_CLASSES 8
#define BATCH 4
#define HW (1024 * 1024)

typedef __attribute__((ext_vector_type(2))) float v2f;
typedef __attribute__((ext_vector_type(8))) float v8f;

// ---------------------------------------------------------------------------
// Zero the 32-entry [B,C] partial-sum table (d_ws is poisoned by the harness).
// ---------------------------------------------------------------------------
__global__ void ce_zero_ws(float* __restrict__ ws) {
    if (threadIdx.x < BATCH * C_CLASSES) ws[threadIdx.x] = 0.0f;
}

// ---------------------------------------------------------------------------
// Streaming pass: for each pixel, log-softmax over C=8, accumulate t*x per
// (b,c).  Bandwidth-bound: 16 coalesced B128 streams per wave, grid-stride
// with global_prefetch_b8 for the next iteration.
// ---------------------------------------------------------------------------
__global__ __launch_bounds__(256) void ce_partial(const float* __restrict__ outs,
                                                  const float* __restrict__ tgts,
                                                  const int* __restrict__ logits_flag,
                                                  float* __restrict__ ws) {
    const int b      = blockIdx.y;
    const int nvec   = HW / 4;                       // float4 groups per plane
    const int stride = gridDim.x * blockDim.x;
    const int lg     = *logits_flag;                 // scalar (SMEM) load

    const float* obase[C_CLASSES];
    const float* tbase[C_CLASSES];
#pragma unroll
    for (int c = 0; c < C_CLASSES; ++c) {
        size_t plane = ((size_t)b * C_CLASSES + c) * (size_t)HW;
        obase[c] = outs + plane;
        tbase[c] = tgts + plane;
    }

    float acc[C_CLASSES];
#pragma unroll
    for (int c = 0; c < C_CLASSES; ++c) acc[c] = 0.0f;

    for (int g = blockIdx.x * blockDim.x + threadIdx.x; g < nvec; g += stride) {
        float4 o[C_CLASSES], t[C_CLASSES];
        const bool pf = (g + stride) < nvec;
#pragma unroll
        for (int c = 0; c < C_CLASSES; ++c) {
            const float4* op = (const float4*)obase[c] + g;
            const float4* tp = (const float4*)tbase[c] + g;
            o[c] = *op;
            t[c] = *tp;
            if (pf) {                                 // -> global_prefetch_b8
                __builtin_prefetch((const void*)(op + stride), 0, 1);
                __builtin_prefetch((const void*)(tp + stride), 0, 1);
            }
        }
#pragma unroll
        for (int j = 0; j < 4; ++j) {
            float xv[C_CLASSES];
#pragma unroll
            for (int c = 0; c < C_CLASSES; ++c) xv[c] = ((const float*)&o[c])[j];
            if (lg) {                                 // wave-uniform branch
                float m = xv[0];
#pragma unroll
                for (int c = 1; c < C_CLASSES; ++c) m = fmaxf(m, xv[c]);
                float se = 0.0f;
#pragma unroll
                for (int c = 0; c < C_CLASSES; ++c) se += __expf(xv[c] - m);
                const float lse = m + __logf(se);
#pragma unroll
                for (int c = 0; c < C_CLASSES; ++c) xv[c] -= lse;
            }
#pragma unroll
            for (int c = 0; c < C_CLASSES; ++c)
                acc[c] = fmaf(((const float*)&t[c])[j], xv[c], acc[c]);
        }
    }

    // wave32 tree reduction, then LDS combine, then one atomic per (block,c)
    __shared__ float sdata[C_CLASSES];
    if (threadIdx.x < C_CLASSES) sdata[threadIdx.x] = 0.0f;
    __syncthreads();
#pragma unroll
    for (int c = 0; c < C_CLASSES; ++c) {
        float v = acc[c];
#pragma unroll
        for (int off = 16; off > 0; off >>= 1) v += __shfl_xor(v, off, 32);
        if ((threadIdx.x & 31) == 0) atomicAdd(&sdata[c], v);
    }
    __syncthreads();
    if (threadIdx.x < C_CLASSES)
        atomicAdd(&ws[b * C_CLASSES + threadIdx.x], sdata[threadIdx.x]);
}

// ---------------------------------------------------------------------------
// Finalize: loss_b = nan_to_num( -(1/(C*N)) * sum_c w_c * S[b,c] ), mean over b.
// The weighted row-sum is done with two chained V_WMMA_F32_16X16X4_F32:
//   D = A1 x ones(4x16) ; D = A2 x ones(4x16) + D
// with w_c pre-applied in VALU so B == all-ones (layout-immune).
// A-layout (documented 16x4 f32): vgpr0 = A[m][hi?2:0], vgpr1 = A[m][hi?3:1],
// m = lane&15, hi = lane>=16.  D-layout: lane 0, element r == D[r][0].
// One wave, no divergence before WMMA => EXEC all-1s as required.
// ---------------------------------------------------------------------------
__global__ __launch_bounds__(32) void ce_finalize(const float* __restrict__ ws,
                                                  const float* __restrict__ wgt,
                                                  float* __restrict__ out) {
    const int  lane = threadIdx.x;
    const int  row  = lane & 15;
    const bool hi   = lane >= 16;

    float a0 = 0.0f, a1 = 0.0f, a2 = 0.0f, a3 = 0.0f;
    const int c0 = hi ? 2 : 0;
    const int c1 = hi ? 3 : 1;
    if (row < BATCH) {                        // rows 4..15 of A stay zero
        a0 = ws[row * C_CLASSES + c0]     * wgt[c0];
        a1 = ws[row * C_CLASSES + c1]     * wgt[c1];
        a2 = ws[row * C_CLASSES + c0 + 4] * wgt[c0 + 4];
        a3 = ws[row * C_CLASSES + c1 + 4] * wgt[c1 + 4];
    }

    v2f A1; A1.x = a0; A1.y = a1;
    v2f A2; A2.x = a2; A2.y = a3;
    v2f B1; B1.x = 1.0f; B1.y = 1.0f;         // all-ones B: layout-proof
    v8f d = {};
    d = __builtin_amdgcn_wmma_f32_16x16x4_f32(false, A1, false, B1, (short)0, d,
                                              false, false);
    d = __builtin_amdgcn_wmma_f32_16x16x4_f32(false, A2, false, B1, (short)0, d,
                                              false, false);

    if (lane == 0) {
        const float invCN = 1.0f / ((float)C_CLASSES * (float)HW);
        float loss = 0.0f;
#pragma unroll
        for (int bb = 0; bb < BATCH; ++bb) {
            float lb = -d[bb] * invCN;        // D[bb][0]
            if (!(lb == lb)) lb = 1.0f;       // nan_to_num(nan=1.0)
            loss += lb;
        }
        out[0] = loss * (1.0f / (float)BATCH);
    }
}

// ---------------------------------------------------------------------------
extern "C" void kernel_launch(void* const* d_in, const int* in_sizes, int n_in,
                              void* d_out, int out_size, void* d_ws, size_t ws_size,
                              hipStream_t stream) {
    const float* outs = (const float*)d_in[0];
    const float* tgts = (const float*)d_in[1];
    const float* wgt  = (const float*)d_in[2];
    const int*   flag = (const int*)d_in[3];
    float*       ws   = (float*)d_ws;
    float*       out  = (float*)d_out;

    ce_zero_ws<<<1, 32, 0, stream>>>(ws);

    dim3 grid(256, BATCH);                    // 1024 blocks, 8 waves each
    ce_partial<<<grid, 256, 0, stream>>>(outs, tgts, flag, ws);

    ce_finalize<<<1, 32, 0, stream>>>(ws, wgt, out);
}